// GAT_21827023798632
// MI455X (gfx1250) — compile-verified
//
#include <hip/hip_runtime.h>
#include <math.h>

typedef __attribute__((ext_vector_type(2))) float v2f;
typedef __attribute__((ext_vector_type(8))) float v8f;

#define NEG_SLOPE 0.2f
#define EPS_GN 1e-05f

// ---------- helpers ----------
__device__ __forceinline__ float leaky(float v) {
    return v > 0.0f ? v : NEG_SLOPE * v;
}
// monotonic order-preserving key for float -> unsigned (for atomicMax)
__device__ __forceinline__ unsigned fkey(float f) {
    unsigned u = __float_as_uint(f);
    return (u & 0x80000000u) ? ~u : (u | 0x80000000u);
}
__device__ __forceinline__ float funkey(unsigned k) {
    unsigned u = (k & 0x80000000u) ? (k & 0x7FFFFFFFu) : ~k;
    return __uint_as_float(u);
}

// ---------- layer-1 GEMM (K=3, trivial; bandwidth-bound) ----------
__global__ void gemm_k3(const float* __restrict__ x,   // n x 3
                        const float* __restrict__ Wl,  // 3 x 128
                        const float* __restrict__ Wr,  // 3 x 128
                        float* __restrict__ xl, float* __restrict__ xr, int n) {
    int i = blockIdx.x * blockDim.x + threadIdx.x;
    if (i >= n * 128) return;
    int row = i >> 7, j = i & 127;
    float x0 = x[row * 3 + 0], x1 = x[row * 3 + 1], x2 = x[row * 3 + 2];
    xl[i] = x0 * Wl[j] + x1 * Wl[128 + j] + x2 * Wl[256 + j];
    xr[i] = x0 * Wr[j] + x1 * Wr[128 + j] + x2 * Wr[256 + j];
}

// ---------- fp32 WMMA GEMM: Y[n x 128] = X[n x 64] @ W[64 x 128] ----------
// One wave computes a 16x64 block (4 accumulator tiles), reusing each A
// fragment for 4 consecutive V_WMMA_F32_16X16X4_F32 issues. 64 WMMA / wave.
__global__ void gemm_wmma_f32(const float* __restrict__ X,
                              const float* __restrict__ W,
                              float* __restrict__ Y, int nrows) {
    const int K = 64, NC = 128;
    int wave = (blockIdx.x * blockDim.x + threadIdx.x) >> 5;
    int lane = threadIdx.x & 31;
    int mt = wave >> 1;                 // 16-row tile index
    int ng = wave & 1;                  // column group: cols [ng*64, ng*64+64)
    if (mt * 16 >= nrows) return;
    int m0 = mt * 16, n0 = ng * 64;
    int half = lane >> 4;               // ISA 16x4 f32 A layout: K base + 2*half
    int m = lane & 15;                  // A row / B,C,D column within tile
    v8f acc0 = {}, acc1 = {}, acc2 = {}, acc3 = {};
    const float* Xr = X + (size_t)(m0 + m) * K;
#pragma unroll
    for (int kk = 0; kk < K; kk += 4) {
        v2f a;
        a.x = Xr[kk + 2 * half];        // VGPR0: K = kk + 2*half
        a.y = Xr[kk + 2 * half + 1];    // VGPR1: K = kk + 2*half + 1
        const float* W0 = W + (size_t)(kk + 2 * half) * NC + n0 + m;
        const float* W1 = W0 + NC;
        v2f b0 = { W0[0],  W1[0]  };
        v2f b1 = { W0[16], W1[16] };
        v2f b2 = { W0[32], W1[32] };
        v2f b3 = { W0[48], W1[48] };
        acc0 = __builtin_amdgcn_wmma_f32_16x16x4_f32(false, a, false, b0, (short)0, acc0, false, false);
        acc1 = __builtin_amdgcn_wmma_f32_16x16x4_f32(false, a, false, b1, (short)0, acc1, false, false);
        acc2 = __builtin_amdgcn_wmma_f32_16x16x4_f32(false, a, false, b2, (short)0, acc2, false, false);
        acc3 = __builtin_amdgcn_wmma_f32_16x16x4_f32(false, a, false, b3, (short)0, acc3, false, false);
    }
    float* Yb = Y + (size_t)m0 * NC + n0 + m;
#pragma unroll
    for (int r = 0; r < 8; ++r) {       // C/D layout: VGPR r -> rows r, r+8
        size_t row = (size_t)(r + 8 * half) * NC;
        Yb[row + 0]  = acc0[r];
        Yb[row + 16] = acc1[r];
        Yb[row + 32] = acc2[r];
        Yb[row + 48] = acc3[r];
    }
}

// ---------- edge phase 1: logits + segment-max ----------
// One wave per edge. lane*4 channels; head0 = lanes 0-15, head1 = lanes 16-31.
__global__ void edge_logit(const float* __restrict__ xl, const float* __restrict__ xr,
                           const float* __restrict__ att,  // 2 x 64
                           const int* __restrict__ ei, int E, int Etot,
                           float* __restrict__ logit, unsigned* __restrict__ mkey) {
    int wave = (blockIdx.x * blockDim.x + threadIdx.x) >> 5;
    int lane = threadIdx.x & 31;
    if (wave >= Etot) return;
    int e = wave;
    int s = (e < E) ? ei[e] : (e - E);
    int d = (e < E) ? ei[E + e] : (e - E);
    float4 l4 = ((const float4*)(xl + (size_t)s * 128))[lane];
    float4 r4 = ((const float4*)(xr + (size_t)d * 128))[lane];
    float4 a4 = ((const float4*)att)[lane];
    float p = leaky(l4.x + r4.x) * a4.x + leaky(l4.y + r4.y) * a4.y +
              leaky(l4.z + r4.z) * a4.z + leaky(l4.w + r4.w) * a4.w;
#pragma unroll
    for (int off = 8; off >= 1; off >>= 1)          // reduce within each 16-lane head
        p += __shfl_xor(p, off, 32);
    if ((lane & 15) == 0) {
        int h = lane >> 4;
        logit[(size_t)e * 2 + h] = p;
        atomicMax(&mkey[(size_t)d * 2 + h], fkey(p));
    }
}

// ---------- edge phase 2: a = exp(logit - m[dst]); z[dst] += a ----------
__global__ void edge_expsum(const float* __restrict__ logit,
                            const unsigned* __restrict__ mkey,
                            const int* __restrict__ ei, int E, int Etot,
                            float* __restrict__ aout, float* __restrict__ z) {
    int e = blockIdx.x * blockDim.x + threadIdx.x;
    if (e >= Etot) return;
    int d = (e < E) ? ei[E + e] : (e - E);
    float m0 = funkey(mkey[(size_t)d * 2 + 0]);
    float m1 = funkey(mkey[(size_t)d * 2 + 1]);
    float a0 = __expf(logit[(size_t)e * 2 + 0] - m0);
    float a1 = __expf(logit[(size_t)e * 2 + 1] - m1);
    aout[(size_t)e * 2 + 0] = a0;
    aout[(size_t)e * 2 + 1] = a1;
    atomicAdd(&z[(size_t)d * 2 + 0], a0);
    atomicAdd(&z[(size_t)d * 2 + 1], a1);
}

// ---------- edge phase 3: acc[dst] += alpha * xl[src] ----------
__global__ void edge_scatter(const float* __restrict__ xl,
                             const float* __restrict__ aout, const float* __restrict__ z,
                             const int* __restrict__ ei, int E, int Etot,
                             float* __restrict__ acc) {
    int wave = (blockIdx.x * blockDim.x + threadIdx.x) >> 5;
    int lane = threadIdx.x & 31;
    if (wave >= Etot) return;
    int e = wave;
    int s = (e < E) ? ei[e] : (e - E);
    int d = (e < E) ? ei[E + e] : (e - E);
    int h = lane >> 4;
    float alpha = aout[(size_t)e * 2 + h] / z[(size_t)d * 2 + h];
    float4 l4 = ((const float4*)(xl + (size_t)s * 128))[lane];
    float* dp = acc + (size_t)d * 128 + lane * 4;
    atomicAdd(dp + 0, alpha * l4.x);
    atomicAdd(dp + 1, alpha * l4.y);
    atomicAdd(dp + 2, alpha * l4.z);
    atomicAdd(dp + 3, alpha * l4.w);
}

// ---------- head-mean + bias, column sum/sumsq (LDS block reduction) ----------
__global__ void finalize_stats(const float* __restrict__ acc, const float* __restrict__ b,
                               float* __restrict__ y, float* __restrict__ stats, int n) {
    int c = threadIdx.x & 63;
    int rg = threadIdx.x >> 6;  // 4 row-groups of 64 threads
    float s = 0.0f, sq = 0.0f;
    for (int row = blockIdx.x * 4 + rg; row < n; row += gridDim.x * 4) {
        float v = 0.5f * (acc[(size_t)row * 128 + c] + acc[(size_t)row * 128 + 64 + c]) + b[c];
        y[(size_t)row * 64 + c] = v;
        s += v;
        sq += v * v;
    }
    __shared__ float shs[4][64];
    __shared__ float shq[4][64];
    shs[rg][c] = s;
    shq[rg][c] = sq;
    __syncthreads();
    if (rg == 0) {
        s  = shs[0][c] + shs[1][c] + shs[2][c] + shs[3][c];
        sq = shq[0][c] + shq[1][c] + shq[2][c] + shq[3][c];
        atomicAdd(&stats[c], s);
        atomicAdd(&stats[64 + c], sq);
    }
}

// ---------- per-column scale/shift: one-pass var ----------
// o = y - gm*mu; E[o^2] = E[y^2] - mu^2 * gm * (2 - gm)
__global__ void norm_coeffs(float* __restrict__ stats,
                            const float* __restrict__ gw, const float* __restrict__ gb,
                            const float* __restrict__ gm, int n) {
    int c = threadIdx.x;
    if (c >= 64) return;
    float inv = 1.0f / (float)n;
    float mu = stats[c] * inv;
    float msq = stats[64 + c] * inv;
    float g = gm[c];
    float var = msq - mu * mu * g * (2.0f - g);
    float scale = gw[c] * rsqrtf(var + EPS_GN);
    stats[128 + c] = scale;
    stats[192 + c] = gb[c] - scale * g * mu;
}

// ---------- apply norm + ReLU ----------
__global__ void normalize_relu(const float* __restrict__ y, const float* __restrict__ stats,
                               float* __restrict__ out, int total) {
    int i = blockIdx.x * blockDim.x + threadIdx.x;
    if (i >= total) return;
    int c = i & 63;
    float v = stats[128 + c] * y[i] + stats[192 + c];
    out[i] = fmaxf(v, 0.0f);
}

extern "C" void kernel_launch(void* const* d_in, const int* in_sizes, int n_in,
                              void* d_out, int out_size, void* d_ws, size_t ws_size,
                              hipStream_t stream) {
    const float* x  = (const float*)d_in[0];
    const int*   ei = (const int*)d_in[1];
    const int N = in_sizes[0] / 3;
    const int E = in_sizes[1] / 2;
    const int Etot = E + N;

    const float *Wl[3], *Wr[3], *att[3], *bb[3], *gw[3], *gb[3], *gm[3];
    for (int L = 0; L < 3; ++L) {
        int base = 2 + 7 * L;
        Wl[L]  = (const float*)d_in[base + 0];
        Wr[L]  = (const float*)d_in[base + 1];
        att[L] = (const float*)d_in[base + 2];
        bb[L]  = (const float*)d_in[base + 3];
        gw[L]  = (const float*)d_in[base + 4];
        gb[L]  = (const float*)d_in[base + 5];
        gm[L]  = (const float*)d_in[base + 6];
    }

    // workspace layout (floats)
    const size_t NM = (size_t)N * 128;
    float* ws   = (float*)d_ws;
    float* xl    = ws;
    float* xr    = ws + NM;
    float* acc   = ws + 2 * NM;
    float* logit = ws + 3 * NM;
    float* aout  = logit + (size_t)2 * Etot;
    unsigned* mkey = (unsigned*)(aout + (size_t)2 * Etot);
    float* z     = (float*)(mkey + (size_t)2 * N);
    float* y     = z + (size_t)2 * N;
    float* xbuf  = y + (size_t)N * 64;
    float* stats = xbuf + (size_t)N * 64;

    const int TPB = 256;
    const int gemm_waves = ((N + 15) / 16) * 2;                   // 16x64 block per wave
    const int gemm_blocks = (gemm_waves * 32 + TPB - 1) / TPB;
    const int ewave_blocks = (Etot + 7) / 8;                      // 8 waves / block
    const int ethr_blocks = (Etot + TPB - 1) / TPB;

    for (int L = 0; L < 3; ++L) {
        hipMemsetAsync(mkey, 0, (size_t)2 * N * sizeof(unsigned), stream);  // key 0 == -inf
        hipMemsetAsync(z, 0, (size_t)2 * N * sizeof(float), stream);
        hipMemsetAsync(acc, 0, NM * sizeof(float), stream);
        hipMemsetAsync(stats, 0, 256 * sizeof(float), stream);

        if (L == 0) {
            gemm_k3<<<(N * 128 + TPB - 1) / TPB, TPB, 0, stream>>>(x, Wl[0], Wr[0], xl, xr, N);
        } else {
            gemm_wmma_f32<<<gemm_blocks, TPB, 0, stream>>>(xbuf, Wl[L], xl, N);
            gemm_wmma_f32<<<gemm_blocks, TPB, 0, stream>>>(xbuf, Wr[L], xr, N);
        }

        edge_logit<<<ewave_blocks, TPB, 0, stream>>>(xl, xr, att[L], ei, E, Etot, logit, mkey);
        edge_expsum<<<ethr_blocks, TPB, 0, stream>>>(logit, mkey, ei, E, Etot, aout, z);
        edge_scatter<<<ewave_blocks, TPB, 0, stream>>>(xl, aout, z, ei, E, Etot, acc);

        finalize_stats<<<512, TPB, 0, stream>>>(acc, bb[L], y, stats, N);
        norm_coeffs<<<1, 64, 0, stream>>>(stats, gw[L], gb[L], gm[L], N);

        float* target = (L == 2) ? (float*)d_out : xbuf;
        normalize_relu<<<(N * 64 + TPB - 1) / TPB, TPB, 0, stream>>>(y, stats, target, N * 64);
    }
}